// RGCN_23313082483289
// MI455X (gfx1250) — compile-verified
//
#include <hip/hip_runtime.h>

// RGCN basis-decomposed conv for MI455X (gfx1250, wave32, WMMA f32 16x16x4).
// D = 100 (hardcoded in the reference), B = 4 bases, R = 40 relations.

typedef float v2f __attribute__((ext_vector_type(2)));
typedef float v8f __attribute__((ext_vector_type(8)));

#define D_DIM 100
#define D_PAD 112            // 7 tiles of 16
#define LDS_STRIDE 104       // 100 cols padded (keeps 8B alignment of rows)

// ---------------------------------------------------------------------------
// x[i,:] = emb[entity[i],:]
__global__ void gather_rows(const float* __restrict__ emb,
                            const int* __restrict__ idx,
                            float* __restrict__ x, int N) {
  long long g = (long long)blockIdx.x * blockDim.x + threadIdx.x;
  if (g >= (long long)N * D_DIM) return;
  int i = (int)(g / D_DIM);
  int d = (int)(g - (long long)i * D_DIM);
  x[g] = emb[(size_t)idx[i] * D_DIM + d];
}

// ---------------------------------------------------------------------------
// Pack all NB basis weights + root into Wt5[w][c][k] = W_w[k][c], zero-padded
// to 112x112 each. Every B-fragment load in the GEMM is then an unconditional
// 8B-aligned global_load_b64 (no exec masking, contiguous K pairs).
__global__ void pad_transpose_all(const float* __restrict__ basis,  // [NB,100,100]
                                  const float* __restrict__ root,   // [100,100]
                                  float* __restrict__ Wt5, int NB) {
  int g = blockIdx.x * blockDim.x + threadIdx.x;
  int total = (NB + 1) * D_PAD * D_PAD;
  if (g >= total) return;
  int w = g / (D_PAD * D_PAD);
  int rem = g - w * (D_PAD * D_PAD);
  int c = rem / D_PAD;
  int k = rem - c * D_PAD;
  const float* W = (w < NB) ? basis + (size_t)w * D_DIM * D_DIM : root;
  Wt5[g] = (c < D_DIM && k < D_DIM) ? W[(size_t)k * D_DIM + c] : 0.0f;
}

// ---------------------------------------------------------------------------
// Fused conv GEMMs: for one 16-row stripe of X (staged in LDS once),
//   H[row, w, :]   = X[row,:] @ basis[w]   for w in 0..NB-1   (interleaved!)
//   Out[row, :]    = X[row,:] @ root + bias
// 7 waves = 7 column tiles; each wave loops over the NB+1 weight matrices.
__global__ __launch_bounds__(224)
void wmma_gemm_fused(const float* __restrict__ X,    // [N, 100]
                     const float* __restrict__ Wt5,  // [NB+1, 112, 112]
                     const float* __restrict__ bias, // [100]
                     float* __restrict__ H,          // [N, NB, 100]
                     float* __restrict__ Out,        // [N, 100]
                     int N, int NB) {
  __shared__ float Xs[16 * LDS_STRIDE];

  const int row0 = blockIdx.x * 16;

  // Cooperative stage of the 16 x 100 A-tile into LDS (read X exactly once
  // for all NB+1 GEMMs).
  for (int i = threadIdx.x; i < 16 * D_DIM; i += blockDim.x) {
    int r = i / D_DIM;
    int c = i - r * D_DIM;
    int gr = row0 + r;
    Xs[r * LDS_STRIDE + c] = (gr < N) ? X[(size_t)gr * D_DIM + c] : 0.0f;
  }
  __syncthreads();

  const int wave = threadIdx.x >> 5;      // column tile id: 0..6
  const int lane = threadIdx.x & 31;
  const int col0 = wave * 16;

  const int hl = lane & 15;               // row (A) / col (B,C) within tile
  const int kk = (lane >> 4) << 1;        // K sub-offset: 0 or 2
  const int bCol = col0 + hl;             // 0..111, always < D_PAD
  const bool cOk = bCol < D_DIM;
  const int rbase = row0 + ((lane >> 4) << 3);
  const float* __restrict__ ap = Xs + hl * LDS_STRIDE;  // this lane's A row

  const int nw = NB + 1;
  const size_t hstride = (size_t)NB * D_DIM;            // H row stride

  for (int w = 0; w < nw; ++w) {
    const float* __restrict__ wp =
        Wt5 + ((size_t)w * D_PAD + bCol) * D_PAD;       // this lane's B column

    v8f acc = {0.f, 0.f, 0.f, 0.f, 0.f, 0.f, 0.f, 0.f};
#pragma unroll
    for (int k = 0; k < D_DIM; k += 4) {
      v2f a = *(const v2f*)(ap + k + kk);   // ds_load_b64 (2addr pairs)
      v2f b = *(const v2f*)(wp + k + kk);   // global_load_b64, L2-resident
      acc = __builtin_amdgcn_wmma_f32_16x16x4_f32(false, a, false, b,
                                                  (short)0, acc, false, false);
    }

    // C/D layout: VGPR r -> M = r (+8 for lanes 16-31), N = lane & 15.
    if (cOk) {
      if (w < NB) {
        float* __restrict__ op = H + (size_t)w * D_DIM + bCol;
#pragma unroll
        for (int r = 0; r < 8; ++r) {
          int gr = rbase + r;
          if (gr < N) op[(size_t)gr * hstride] = acc[r];
        }
      } else {
        const float bv = bias[bCol];
#pragma unroll
        for (int r = 0; r < 8; ++r) {
          int gr = rbase + r;
          if (gr < N) Out[(size_t)gr * D_DIM + bCol] = acc[r] + bv;
        }
      }
    }
  }
}

// ---------------------------------------------------------------------------
// msg_e = norm_e * sum_b comb[type_e,b] * H[src_e, b, :]; atomic add at dst_e.
// H interleaved [N, NB, 100]: the 4 gathers per (edge,d) hit one contiguous
// 1.6 KB region -> few cachelines, L2-resident (H ~160 MB < 192 MB L2).
// 128 threads per edge (d contiguous within each wave32 -> coalesced).
__global__ void edge_scatter(const int* __restrict__ src,
                             const int* __restrict__ dst,
                             const int* __restrict__ etype,
                             const float* __restrict__ enorm,
                             const float* __restrict__ comb,   // [R, NB]
                             const float* __restrict__ H,      // [N, NB, 100]
                             float* __restrict__ out,          // [N, 100]
                             int E, int NB) {
  long long g = (long long)blockIdx.x * blockDim.x + threadIdx.x;
  int e = (int)(g >> 7);
  int d = (int)(g & 127);
  if (e >= E || d >= D_DIM) return;

  const float nrm = enorm[e];
  const float* cb = comb + (size_t)etype[e] * NB;
  const float* hp = H + (size_t)src[e] * NB * D_DIM + d;

  float v = 0.0f;
#pragma unroll 4
  for (int b = 0; b < NB; ++b) v += cb[b] * hp[(size_t)b * D_DIM];

  atomicAdd(&out[(size_t)dst[e] * D_DIM + d], nrm * v);
}

// ---------------------------------------------------------------------------
__global__ void relu_inplace(float* __restrict__ x, long long n) {
  long long g = (long long)blockIdx.x * blockDim.x + threadIdx.x;
  if (g < n) x[g] = fmaxf(x[g], 0.0f);
}

// ---------------------------------------------------------------------------
static void run_conv(const float* x, const float* basis, const float* comb,
                     const float* root, const float* bias, float* out,
                     float* H, float* Wt5, const int* src, const int* dst,
                     const int* etype, const float* enorm,
                     int N, int E, int NB, int do_relu, hipStream_t stream) {
  // Pack + pad + transpose all NB+1 weight matrices (tiny, L2-resident).
  int pad_total = (NB + 1) * D_PAD * D_PAD;
  pad_transpose_all<<<(pad_total + 255) / 256, 256, 0, stream>>>(basis, root,
                                                                 Wt5, NB);

  // H (interleaved) and out = x @ root + bias in one fused pass over X.
  wmma_gemm_fused<<<(N + 15) / 16, 224, 0, stream>>>(x, Wt5, bias, H, out, N,
                                                     NB);

  // out += segment_sum over edges
  long long etot = (long long)E * 128;
  edge_scatter<<<(int)((etot + 255) / 256), 256, 0, stream>>>(
      src, dst, etype, enorm, comb, H, out, E, NB);

  if (do_relu) {
    long long tot = (long long)N * D_DIM;
    relu_inplace<<<(int)((tot + 255) / 256), 256, 0, stream>>>(out, tot);
  }
}

// ---------------------------------------------------------------------------
extern "C" void kernel_launch(void* const* d_in, const int* in_sizes, int n_in,
                              void* d_out, int out_size, void* d_ws,
                              size_t ws_size, hipStream_t stream) {
  const int*   entity   = (const int*)d_in[0];
  const int*   edge_idx = (const int*)d_in[1];     // [2, E]
  const int*   etype    = (const int*)d_in[2];
  const float* enorm    = (const float*)d_in[3];
  const float* emb      = (const float*)d_in[4];
  const float* basis1   = (const float*)d_in[5];
  const float* comb1    = (const float*)d_in[6];
  const float* root1    = (const float*)d_in[7];
  const float* bias1    = (const float*)d_in[8];
  const float* basis2   = (const float*)d_in[9];
  const float* comb2    = (const float*)d_in[10];
  const float* root2    = (const float*)d_in[11];
  const float* bias2    = (const float*)d_in[12];

  const int N  = in_sizes[0];
  const int E  = in_sizes[2];
  const int NB = in_sizes[5] / (D_DIM * D_DIM);    // 4 bases

  float* bufA = (float*)d_ws;                      // [N, 100]
  float* bufB = bufA + (size_t)N * D_DIM;          // [N, 100]
  float* H    = bufB + (size_t)N * D_DIM;          // [N, NB, 100] (~160 MB: L2-resident)
  float* Wt5  = H + (size_t)NB * N * D_DIM;        // [NB+1, 112, 112] packed weights

  const int* src = edge_idx;
  const int* dst = edge_idx + E;

  // x0 = emb[entity]
  {
    long long tot = (long long)N * D_DIM;
    gather_rows<<<(int)((tot + 255) / 256), 256, 0, stream>>>(emb, entity,
                                                              bufA, N);
  }

  // conv1, conv1+relu, conv2 (reference applies conv1 twice)
  run_conv(bufA, basis1, comb1, root1, bias1, bufB, H, Wt5, src, dst, etype,
           enorm, N, E, NB, 0, stream);
  run_conv(bufB, basis1, comb1, root1, bias1, bufA, H, Wt5, src, dst, etype,
           enorm, N, E, NB, 1, stream);
  run_conv(bufA, basis2, comb2, root2, bias2, (float*)d_out, H, Wt5, src, dst,
           etype, enorm, N, E, NB, 0, stream);
}